// PatchMMD_50362786512982
// MI455X (gfx1250) — compile-verified
//
#include <hip/hip_runtime.h>
#include <math.h>

#define PATCH   7
#define IMG     48
#define CH      3
#define OUTW    42                 // 48-7+1
#define NPATCH  (OUTW*OUTW)        // 1764
#define T_REAL  (2*NPATCH)         // 3528
#define T_PAD   3584               // 112 blocks of 32; zero-padded rows (s=0)
#define NB32    (T_PAD/32)         // 112
#define NTRI    (NB32*(NB32+1)/2)  // 6328 upper-tri 32x32 blocks (divisible by 8)
#define D_REAL  147                // 3*7*7
#define LDK     148                // padded K (col 147 = 0)
#define BATCH   4

typedef __attribute__((ext_vector_type(2))) float v2f;
typedef __attribute__((ext_vector_type(8))) float v8f;

// ---------------------------------------------------------------- helpers ---
__device__ __forceinline__ float fast_exp2(float x) {
#if __has_builtin(__builtin_amdgcn_exp2f)
  return __builtin_amdgcn_exp2f(x);    // v_exp_f32
#else
  return exp2f(x);
#endif
}

// sum over the 6 GMMN bandwidths of exp(e/sigma), e = -0.5*||pi-pj||^2 <= 0.
// sigmas {2,5,10,20,40,80} chain from 80: one v_exp_f32 + 6 v_mul_f32.
__device__ __forceinline__ float kern6(float e) {
  const float L2E = 1.4426950408889634f;
  float u   = fast_exp2(e * (L2E / 80.0f));  // exp(e/80)
  float u2  = u   * u;                       // exp(e/40)
  float u4  = u2  * u2;                      // exp(e/20)
  float u8  = u4  * u4;                      // exp(e/10)
  float u16 = u8  * u8;                      // exp(e/5)
  float u32 = u16 * u16;
  float u40 = u32 * u8;                      // exp(e/2)
  return u + u2 + u4 + u8 + u16 + u40;
}

__device__ __forceinline__ float svec(int r) {
  const float inv = 1.0f / (float)NPATCH;
  return (r < NPATCH) ? inv : ((r < T_REAL) ? -inv : 0.0f);
}

__device__ __forceinline__ v8f wmma_f32(v2f a, v2f b, v8f c) {
  // V_WMMA_F32_16X16X4_F32 : D = A(16x4) * B(4x16) + C
  return __builtin_amdgcn_wmma_f32_16x16x4_f32(
      /*neg_a=*/false, a, /*neg_b=*/false, b,
      /*c_mod=*/(short)0, c, /*reuse_a=*/false, /*reuse_b=*/false);
}

// Decode linear index t into upper-triangle (I<=J) block coords, row-major.
// base(I) = I*nb - I*(I-1)/2 ; wave-uniform SALU work.
__device__ __forceinline__ void tri_decode(int t, int nb, int& I, int& J) {
  float nf = (float)nb + 0.5f;
  int i = (int)(nf - sqrtf(fmaxf(nf * nf - 2.0f * (float)t, 0.0f)));
  if (i < 0) i = 0;
  if (i > nb - 1) i = nb - 1;
  while (i > 0 && i * nb - i * (i - 1) / 2 > t) --i;
  while ((i + 1) * nb - (i + 1) * i / 2 <= t) ++i;
  I = i;
  J = i + (t - (i * nb - i * (i - 1) / 2));
}

// ------------------------------------------------------------- extraction ---
// P[b][row][k] row-major with LDK pad; rows 0..1763 from x, 1764..3527 from y,
// rest zero. k 0..146 valid, col 147 zero.
__global__ void mmd_extract(const float* __restrict__ x,
                            const float* __restrict__ y,
                            float* __restrict__ P) {
  long idx = (long)blockIdx.x * blockDim.x + threadIdx.x;
  const long total = (long)BATCH * T_PAD * LDK;
  if (idx >= total) return;
  int k = (int)(idx % LDK);
  int r = (int)((idx / LDK) % T_PAD);
  int b = (int)(idx / ((long)LDK * T_PAD));
  float v = 0.0f;
  if (k < D_REAL && r < T_REAL) {
    const float* im = (r < NPATCH) ? x : y;
    int p  = (r < NPATCH) ? r : r - NPATCH;
    int ph = p / OUTW, pw = p % OUTW;
    int c   = k / (PATCH * PATCH);
    int rem = k % (PATCH * PATCH);
    int py  = rem / PATCH, px = rem % PATCH;
    v = im[(((long)b * CH + c) * IMG + (ph + py)) * IMG + (pw + px)];
  }
  P[idx] = v;
}

// ----------------------------------------------------------------- norms ----
__global__ void mmd_sqnorm(const float* __restrict__ P, float* __restrict__ sq) {
  int idx = blockIdx.x * blockDim.x + threadIdx.x;
  if (idx >= BATCH * T_PAD) return;
  const float* row = P + (long)idx * LDK;
  float s = 0.0f;
  #pragma unroll 4
  for (int k = 0; k < LDK; k += 4) {
    float4 v = *(const float4*)(row + k);
    s += v.x * v.x + v.y * v.y + v.z * v.z + v.w * v.w;
  }
  sq[idx] = s;
}

__global__ void mmd_zero(float* __restrict__ loss) {
  if (threadIdx.x < BATCH) loss[threadIdx.x] = 0.0f;
}

// ------------------------------------------------------------- main fused ---
// grid = (NTRI/8, 1, BATCH), block = 256 (8 waves). Each wave owns one 32x32
// upper-triangle block (2x2 WMMA tiles): off-diagonal blocks weighted 2x by
// symmetry of s_i s_j k(p_i, p_j); diagonal blocks weighted 1x.
__global__ __launch_bounds__(256) void mmd_main(const float* __restrict__ P,
                                                const float* __restrict__ sq,
                                                float* __restrict__ loss) {
  const int b = blockIdx.z;
  const float* Pb  = P  + (long)b * T_PAD * LDK;
  const float* sqb = sq + (long)b * T_PAD;

  const int lane = threadIdx.x & 31;
  const int wave = threadIdx.x >> 5;            // 0..7
  const int t    = blockIdx.x * 8 + wave;       // < NTRI exactly

  int I, J;
  tri_decode(t, NB32, I, J);
  const int rowBase = I * 32;
  const int colBase = J * 32;
  const float wgt = (I == J) ? 1.0f : 2.0f;

  const int half = lane >> 4;                   // K-half select (fp32 A/B frag)
  const int m    = lane & 15;                   // M (A) / N (B) within tile

  // A frags: row tiles rowBase, rowBase+16. B frags: col tiles (B = P^T ->
  // identical load pattern on the row-major patch matrix).
  const float* a0p = Pb + (long)(rowBase + m) * LDK + 2 * half;
  const float* a1p = a0p + 16 * LDK;
  const float* b0p = Pb + (long)(colBase + m) * LDK + 2 * half;
  const float* b1p = b0p + 16 * LDK;

  v8f c00 = {}, c01 = {}, c10 = {}, c11 = {};
  for (int k = 0; k < LDK; k += 4) {            // 37 steps, 148 WMMAs
    v2f a0 = *(const v2f*)(a0p + k);
    v2f a1 = *(const v2f*)(a1p + k);
    v2f b0 = *(const v2f*)(b0p + k);
    v2f b1 = *(const v2f*)(b1p + k);
    c00 = wmma_f32(a0, b0, c00);
    c01 = wmma_f32(a0, b1, c01);
    c10 = wmma_f32(a1, b0, c10);
    c11 = wmma_f32(a1, b1, c11);
  }

  // Epilogue: C/D layout -> VGPR e, lanes 0-15: M=e,N=lane; 16-31: M=e+8.
  const int col0 = colBase + m;
  const int col1 = col0 + 16;
  const float sc0 = svec(col0), sc1 = svec(col1);
  const float qc0 = sqb[col0],  qc1 = sqb[col1];

  float acc = 0.0f;
  #pragma unroll
  for (int e = 0; e < 8; ++e) {
    const int row0 = rowBase + e + 8 * half;
    const int row1 = row0 + 16;
    const float sr0 = svec(row0), sr1 = svec(row1);
    const float qr0 = sqb[row0],  qr1 = sqb[row1];
    acc += kern6(c00[e] - 0.5f * (qr0 + qc0)) * (sr0 * sc0);
    acc += kern6(c01[e] - 0.5f * (qr0 + qc1)) * (sr0 * sc1);
    acc += kern6(c10[e] - 0.5f * (qr1 + qc0)) * (sr1 * sc0);
    acc += kern6(c11[e] - 0.5f * (qr1 + qc1)) * (sr1 * sc1);
  }
  acc *= wgt;

  // wave32 reduction, then one atomic per wave
  #pragma unroll
  for (int off = 16; off > 0; off >>= 1)
    acc += __shfl_down(acc, off, 32);
  if (lane == 0)
    __hip_atomic_fetch_add(&loss[b], acc, __ATOMIC_RELAXED,
                           __HIP_MEMORY_SCOPE_AGENT);
}

// -------------------------------------------------------------- finalize ----
__global__ void mmd_finalize(const float* __restrict__ loss,
                             float* __restrict__ out) {
  if (threadIdx.x == 0) {
    float s = 0.0f;
    #pragma unroll
    for (int b = 0; b < BATCH; ++b) s += sqrtf(fmaxf(loss[b], 0.0f));
    out[0] = s * (1.0f / BATCH);
  }
}

// ---------------------------------------------------------------- launch ----
extern "C" void kernel_launch(void* const* d_in, const int* in_sizes, int n_in,
                              void* d_out, int out_size, void* d_ws, size_t ws_size,
                              hipStream_t stream) {
  (void)in_sizes; (void)n_in; (void)out_size; (void)ws_size;
  const float* x = (const float*)d_in[0];
  const float* y = (const float*)d_in[1];
  float* out = (float*)d_out;

  float* P    = (float*)d_ws;                         // BATCH*T_PAD*LDK
  float* sq   = P  + (long)BATCH * T_PAD * LDK;       // BATCH*T_PAD
  float* loss = sq + (long)BATCH * T_PAD;             // BATCH

  {
    const long total = (long)BATCH * T_PAD * LDK;
    const int blk = 256;
    const int grd = (int)((total + blk - 1) / blk);
    mmd_extract<<<grd, blk, 0, stream>>>(x, y, P);
  }
  {
    const int total = BATCH * T_PAD;
    mmd_sqnorm<<<(total + 255) / 256, 256, 0, stream>>>(P, sq);
  }
  mmd_zero<<<1, 32, 0, stream>>>(loss);
  {
    dim3 grid(NTRI / 8, 1, BATCH);                    // (791, 1, 4)
    mmd_main<<<grid, 256, 0, stream>>>(P, sq, loss);
  }
  mmd_finalize<<<1, 32, 0, stream>>>(loss, out);
}